// GCN_Net_81243601371599
// MI455X (gfx1250) — compile-verified
//
#include <hip/hip_runtime.h>
#include <hip/hip_bf16.h>
#include <math.h>

typedef float v2f __attribute__((ext_vector_type(2)));
typedef float v8f __attribute__((ext_vector_type(8)));

#define DIM 128

// ---------------- degree / normalization ----------------

__global__ void init_deg_kernel(float* deg, int n) {
    int i = blockIdx.x * blockDim.x + threadIdx.x;
    if (i < n) deg[i] = 1.0f;            // self-loop contribution
}

__global__ void deg_accum_kernel(const int* __restrict__ dst, float* deg, int nE) {
    int e = blockIdx.x * blockDim.x + threadIdx.x;
    if (e < nE) atomicAdd(&deg[dst[e]], 1.0f);
}

__global__ void deg_rsqrt_kernel(float* deg, int n) {
    int i = blockIdx.x * blockDim.x + threadIdx.x;
    if (i < n) deg[i] = rsqrtf(fmaxf(deg[i], 1.0f));   // deg >= 1 always
}

// ---------------- WMMA fp32 GEMM:  Out[N,128] = relu?(A[N,128]) @ W[128,128] ----------------
// Weight-stationary: W is staged transposed in LDS once per block; block then
// sweeps row tiles persistently. All K-loop operand reads are LDS b64 loads.

__global__ void __launch_bounds__(256)
gemm_node_wmma(const float* __restrict__ A, const float* __restrict__ W,
               float* __restrict__ Out, int relu_in, int nN, int nTiles)
{
    __shared__ float Wt[DIM][DIM];   // Wt[n][k] = W[k][n]  (64 KB)
    __shared__ float At[16][DIM];    // current A row-tile  (8 KB)

    const int tid  = threadIdx.x;
    const int wave = tid >> 5;          // 0..7 -> 16-column slab
    const int lane = tid & 31;
    const int m    = lane & 15;         // row within tile (A) / col within slab (B,C)
    const int kh   = lane >> 4;         // half-wave selector
    const int n    = wave * 16 + m;     // output column

    // stage W transposed into LDS once
    for (int i = tid; i < DIM * DIM; i += 256) {
        Wt[i & 127][i >> 7] = W[i];
    }

    const float2* bp = (const float2*)&Wt[n][0];
    const float2* ap = (const float2*)&At[m][0];

    for (int t = blockIdx.x; t < nTiles; t += gridDim.x) {
        const size_t row0 = (size_t)t * 16;

        __syncthreads();   // WAR: previous tile's fragment reads done (also covers Wt init)
        for (int i = tid; i < 16 * DIM; i += 256) {
            size_t r = row0 + (i >> 7);
            float v = (r < (size_t)nN) ? A[r * DIM + (i & 127)] : 0.0f;
            if (relu_in) v = fmaxf(v, 0.0f);
            At[i >> 7][i & 127] = v;
        }
        __syncthreads();   // RAW: tile visible

        v8f c = {};
        #pragma unroll 4
        for (int k0 = 0; k0 < DIM; k0 += 4) {
            // A 16x4: lane m, K = 2*kh,2*kh+1 ; B 4x16 mirrors A (row K striped over lanes)
            const int idx = (k0 >> 1) + kh;     // float2 index: k = k0 + 2*kh
            float2 af = ap[idx];
            float2 bf = bp[idx];
            v2f a; a.x = af.x; a.y = af.y;
            v2f b; b.x = bf.x; b.y = bf.y;
            c = __builtin_amdgcn_wmma_f32_16x16x4_f32(false, a, false, b, (short)0, c,
                                                      false, false);
        }

        // C/D 16x16 f32: VGPR r -> (M = r + 8*kh, N = n)
        float* orow = Out + row0 * DIM + n;
        if (row0 + 16 <= (size_t)nN) {          // fast path: full tile, no predication
            #pragma unroll
            for (int r = 0; r < 8; ++r) {
                orow[(size_t)(r + 8 * kh) * DIM] = c[r];
            }
        } else {
            #pragma unroll
            for (int r = 0; r < 8; ++r) {
                size_t row = row0 + r + 8 * kh;
                if (row < (size_t)nN) Out[row * DIM + n] = c[r];
            }
        }
    }
}

// ---------------- bias init + edge scatter ----------------

__global__ void init_bias_kernel(float* __restrict__ H, const float* __restrict__ b,
                                 long long total) {
    long long i = (long long)blockIdx.x * blockDim.x + threadIdx.x;
    if (i < total) H[i] = b[i & 127];
}

// one wave per (edge or self-loop); lane handles 4 consecutive feature channels
__global__ void __launch_bounds__(256)
scatter_msgs(const float* __restrict__ L, const int* __restrict__ src,
             const int* __restrict__ dst, const float* __restrict__ dinv,
             float* __restrict__ O, int nE, int nN)
{
    const int lane = threadIdx.x & 31;
    long long w = (long long)blockIdx.x * 8 + (threadIdx.x >> 5);
    long long total = (long long)nE + nN;
    if (w >= total) return;

    int s, d; float nm;
    if (w < nE) {
        s = src[w]; d = dst[w];
        nm = dinv[s] * dinv[d];
    } else {
        s = d = (int)(w - nE);
        float di = dinv[s];
        nm = di * di;
    }
    const float4 v = ((const float4*)(L + (size_t)s * DIM))[lane];
    float* op = O + (size_t)d * DIM + lane * 4;
    atomicAdd(op + 0, v.x * nm);
    atomicAdd(op + 1, v.y * nm);
    atomicAdd(op + 2, v.z * nm);
    atomicAdd(op + 3, v.w * nm);
}

// ---------------- mean pool + head ----------------

__global__ void zero_pool_kernel(float* sums, float* cnts, int nSums, int nG) {
    int i = blockIdx.x * blockDim.x + threadIdx.x;
    if (i < nSums) sums[i] = 0.0f;
    if (i < nG)    cnts[i] = 0.0f;
}

__global__ void __launch_bounds__(256)
pool_accum(const float* __restrict__ H, const int* __restrict__ batch,
           float* __restrict__ sums, float* __restrict__ cnts, int nN)
{
    const int lane = threadIdx.x & 31;
    long long i = (long long)blockIdx.x * 8 + (threadIdx.x >> 5);
    if (i >= nN) return;
    int g = batch[i];
    const float4 v = ((const float4*)(H + (size_t)i * DIM))[lane];
    float* sp = sums + (size_t)g * DIM + lane * 4;
    atomicAdd(sp + 0, fmaxf(v.x, 0.0f));   // fused final ReLU
    atomicAdd(sp + 1, fmaxf(v.y, 0.0f));
    atomicAdd(sp + 2, fmaxf(v.z, 0.0f));
    atomicAdd(sp + 3, fmaxf(v.w, 0.0f));
    if (lane == 0) atomicAdd(&cnts[g], 1.0f);
}

__global__ void pool_div_kernel(const float* __restrict__ sums,
                                const float* __restrict__ cnts,
                                float* __restrict__ g, int total) {
    int i = blockIdx.x * blockDim.x + threadIdx.x;
    if (i < total) g[i] = sums[i] / fmaxf(cnts[i >> 7], 1.0f);
}

// one block per graph: 128x128 lin1 + relu + 128x2 lin2 + log_softmax
__global__ void __launch_bounds__(128)
head_kernel(const float* __restrict__ gmean, const float* __restrict__ w1,
            const float* __restrict__ b1, const float* __restrict__ w2,
            const float* __restrict__ b2, float* __restrict__ out)
{
    __shared__ float gv[DIM];
    __shared__ float h1[DIM];
    const int g = blockIdx.x;
    const int j = threadIdx.x;
    gv[j] = gmean[(size_t)g * DIM + j];
    __syncthreads();
    float acc = b1[j];
    for (int k = 0; k < DIM; ++k) acc += gv[k] * w1[(size_t)k * DIM + j];
    h1[j] = fmaxf(acc, 0.0f);
    __syncthreads();
    if (j == 0) {
        float l0 = b2[0], l1 = b2[1];
        for (int k = 0; k < DIM; ++k) {
            l0 += h1[k] * w2[k * 2 + 0];
            l1 += h1[k] * w2[k * 2 + 1];
        }
        float mx  = fmaxf(l0, l1);
        float lse = mx + logf(expf(l0 - mx) + expf(l1 - mx));
        out[g * 2 + 0] = l0 - lse;
        out[g * 2 + 1] = l1 - lse;
    }
}

// ---------------- launch ----------------

extern "C" void kernel_launch(void* const* d_in, const int* in_sizes, int n_in,
                              void* d_out, int out_size, void* d_ws, size_t ws_size,
                              hipStream_t stream) {
    const float* x      = (const float*)d_in[0];
    const int*   ei     = (const int*)  d_in[1];
    const int*   batch  = (const int*)  d_in[2];
    const float* conv_w = (const float*)d_in[3];
    const float* conv_b = (const float*)d_in[4];
    const float* lin1_w = (const float*)d_in[5];
    const float* lin1_b = (const float*)d_in[6];
    const float* lin2_w = (const float*)d_in[7];
    const float* lin2_b = (const float*)d_in[8];
    float* out = (float*)d_out;

    const int N      = in_sizes[0] / DIM;
    const int E      = in_sizes[1] / 2;
    const int NLAYER = in_sizes[3] / (DIM * DIM);
    const int G      = out_size / 2;

    const int* src = ei;
    const int* dst = ei + E;

    // workspace carve-out (256B aligned)
    char* ws = (char*)d_ws;
    size_t off = 0;
    auto carve = [&](size_t bytes) -> void* {
        off = (off + 255) & ~(size_t)255;
        void* p = ws + off;
        off += bytes;
        return p;
    };
    float* dinv  = (float*)carve((size_t)N * 4);
    float* Lbuf  = (float*)carve((size_t)N * DIM * 4);
    float* Hbuf  = (float*)carve((size_t)N * DIM * 4);
    float* sums  = (float*)carve((size_t)G * DIM * 4);
    float* cnts  = (float*)carve((size_t)G * 4);
    float* gmean = (float*)carve((size_t)G * DIM * 4);
    (void)ws_size; (void)n_in;

    // 1) degree -> deg^-1/2
    init_deg_kernel<<<(N + 255) / 256, 256, 0, stream>>>(dinv, N);
    deg_accum_kernel<<<(E + 255) / 256, 256, 0, stream>>>(dst, dinv, E);
    deg_rsqrt_kernel<<<(N + 255) / 256, 256, 0, stream>>>(dinv, N);

    // 2) GCN layers
    const long long totNF = (long long)N * DIM;
    const int nTiles        = (N + 15) / 16;
    const int gemmBlocks    = nTiles < 1280 ? nTiles : 1280;   // weight-stationary sweep
    const int scatterBlocks = (int)(((long long)E + N + 7) / 8);
    const float* cur = x;
    for (int l = 0; l < NLAYER; ++l) {
        gemm_node_wmma<<<gemmBlocks, 256, 0, stream>>>(
            cur, conv_w + (size_t)l * DIM * DIM, Lbuf, l > 0 ? 1 : 0, N, nTiles);
        init_bias_kernel<<<(int)((totNF + 255) / 256), 256, 0, stream>>>(
            Hbuf, conv_b + (size_t)l * DIM, totNF);
        scatter_msgs<<<scatterBlocks, 256, 0, stream>>>(
            Lbuf, src, dst, dinv, Hbuf, E, N);
        cur = Hbuf;   // cur is fully consumed by the GEMM before Hbuf is rewritten
    }

    // 3) global mean pool (final ReLU fused into accumulation)
    zero_pool_kernel<<<(G * DIM + 255) / 256, 256, 0, stream>>>(sums, cnts, G * DIM, G);
    pool_accum<<<(N + 7) / 8, 256, 0, stream>>>(Hbuf, batch, sums, cnts, N);
    pool_div_kernel<<<(G * DIM + 255) / 256, 256, 0, stream>>>(sums, cnts, gmean, G * DIM);

    // 4) MLP head + log_softmax
    head_kernel<<<G, 128, 0, stream>>>(gmean, lin1_w, lin1_b, lin2_w, lin2_b, out);
}